// FSQCodebook_35613868819126
// MI455X (gfx1250) — compile-verified
//
#include <hip/hip_runtime.h>
#include <hip/hip_bf16.h>

typedef float v2f __attribute__((ext_vector_type(2)));
typedef float v8f __attribute__((ext_vector_type(8)));

#define FSQ_B_LATENT   64
#define FSQ_ROWS       256      // rows per block (8 waves x 32 rows)
#define FSQ_LDS_STRIDE 65       // padded row stride (floats) -> conflict-free column reads
#define FSQ_NUM_CODES  1024

// ---------------------------------------------------------------------------
// Kernel 1: zero the global histogram (d_ws scratch must be re-inited each call)
// ---------------------------------------------------------------------------
__global__ void fsq_zero_counts(unsigned* __restrict__ counts) {
    int i = blockIdx.x * blockDim.x + threadIdx.x;
    if (i < FSQ_NUM_CODES) counts[i] = 0u;
}

// ---------------------------------------------------------------------------
// Kernel 2: main streaming kernel
//   - stage 256x64 z_e tile in LDS (read z_e exactly once from HBM)
//   - proj_in via per-lane VALU FMAs (K=64, N=4: too skinny for 16x16 tiles)
//   - FSQ quantize + mixed-radix index + LDS histogram
//   - proj_out via V_WMMA_F32_16X16X4_F32 (exact shape match)
//   - residual z_q = z_e + (z_q_latent - z_e), coalesced stores
// ---------------------------------------------------------------------------
__global__ __launch_bounds__(256) void fsq_main(
    const float* __restrict__ z_e,
    const float* __restrict__ w_in,    // (4, 64) row-major
    const float* __restrict__ w_out,   // (64, 4) row-major
    float* __restrict__ zq_out,        // (B, 64)
    float* __restrict__ idx_out,       // (B,) stored as float
    unsigned* __restrict__ g_counts)   // (1024,)
{
    __shared__ float    s_z[FSQ_ROWS * FSQ_LDS_STRIDE]; // 66560 B
    __shared__ float    s_win[4 * FSQ_B_LATENT];        // 1 KB
    __shared__ float    s_wout[FSQ_B_LATENT * 4];       // 1 KB
    __shared__ float    s_fsq[FSQ_ROWS * 4];            // 4 KB: centered codes
    __shared__ unsigned s_hist[FSQ_NUM_CODES];          // 4 KB

    const int tid = threadIdx.x;
    const long long rowBase = (long long)blockIdx.x * FSQ_ROWS;

    // --- weights into LDS (broadcast reads later), zero LDS histogram ---
    s_win[tid]  = w_in[tid];    // 256 elements each, blockDim == 256
    s_wout[tid] = w_out[tid];
    #pragma unroll
    for (int i = tid; i < FSQ_NUM_CODES; i += 256) s_hist[i] = 0u;

    // --- cooperative, fully-coalesced tile load: 16 float4 per thread ---
    const float4* zsrc = (const float4*)(z_e + rowBase * FSQ_B_LATENT);
    #pragma unroll
    for (int i = 0; i < 16; ++i) {
        int c4   = tid + i * 256;       // float4 chunk id within tile
        int elem = c4 << 2;
        int row  = elem >> 6;           // / 64
        int col  = elem & 63;
        float4 v = zsrc[c4];
        float* dst = &s_z[row * FSQ_LDS_STRIDE + col];
        dst[0] = v.x; dst[1] = v.y; dst[2] = v.z; dst[3] = v.w;
    }
    __syncthreads();

    // --- proj_in + FSQ quantization: one row per thread ---
    {
        float a0 = 0.f, a1 = 0.f, a2 = 0.f, a3 = 0.f;
        const float* zr = &s_z[tid * FSQ_LDS_STRIDE];
        #pragma unroll 16
        for (int j = 0; j < FSQ_B_LATENT; ++j) {
            float zj = zr[j];
            a0 = fmaf(zj, s_win[j],       a0);
            a1 = fmaf(zj, s_win[64 + j],  a1);
            a2 = fmaf(zj, s_win[128 + j], a2);
            a3 = fmaf(zj, s_win[192 + j], a3);
        }
        const float acc[4]    = {a0, a1, a2, a3};
        const float levm1[4]  = {7.f, 7.f, 7.f, 1.f};
        const int   stride[4] = {1, 8, 64, 512};
        int idx = 0;
        #pragma unroll
        for (int k = 0; k < 4; ++k) {
            float u = (tanhf(acc[k]) + 1.f) * 0.5f;   // (tanh+1)/2 in [0,1]
            float s = u * levm1[k];
            float r = rintf(s);                       // round-to-nearest-even
            r = fminf(fmaxf(r, 0.f), levm1[k]);       // clip
            s_fsq[tid * 4 + k] = r - levm1[k] * 0.5f; // centered code
            idx += (int)r * stride[k];
        }
        atomicAdd(&s_hist[idx], 1u);
        idx_out[rowBase + tid] = (float)idx;
    }
    __syncthreads();

    // --- proj_out via WMMA f32 16x16x4, D = A(16x4) x B(4x16) ---
    {
        const int lane = tid & 31;
        const int wave = tid >> 5;
        const int m    = lane & 15;   // row (A) / column (B,D) within tile
        const int kp   = lane >> 4;   // 0 -> K=0,1 ; 1 -> K=2,3

        #pragma unroll
        for (int t = 0; t < 2; ++t) {
            const int base = wave * 32 + t * 16;     // 16-row tile base

            // A-matrix (16x4 f32): lanes 0-15 hold K0/K1, lanes 16-31 hold K2/K3
            v2f A;
            A.x = s_fsq[(base + m) * 4 + 2 * kp];
            A.y = s_fsq[(base + m) * 4 + 2 * kp + 1];

            #pragma unroll
            for (int ct = 0; ct < 4; ++ct) {
                const int n0 = ct * 16;              // latent column base

                // B-matrix (4x16 f32): W_out.T[k][n] = w_out[n*4 + k]
                v2f Bm;
                Bm.x = s_wout[(n0 + m) * 4 + 2 * kp];
                Bm.y = s_wout[(n0 + m) * 4 + 2 * kp + 1];

                v8f C = {0.f, 0.f, 0.f, 0.f, 0.f, 0.f, 0.f, 0.f};
                C = __builtin_amdgcn_wmma_f32_16x16x4_f32(
                        /*neg_a=*/false, A, /*neg_b=*/false, Bm,
                        /*c_mod=*/(short)0, C,
                        /*reuse_a=*/false, /*reuse_b=*/false);

                // D layout: VGPR r = row (base+r) for lanes 0-15, (base+8+r) for 16-31
                #pragma unroll
                for (int r = 0; r < 8; ++r) {
                    int row = base + r + 8 * kp;
                    int col = n0 + m;
                    float ze = s_z[row * FSQ_LDS_STRIDE + col];
                    float zq = ze + (C[r] - ze);     // straight-through forward
                    zq_out[(rowBase + row) * FSQ_B_LATENT + col] = zq;
                }
            }
        }
    }
    __syncthreads();

    // --- flush per-block histogram to global counts ---
    #pragma unroll
    for (int i = tid; i < FSQ_NUM_CODES; i += 256) {
        unsigned c = s_hist[i];
        if (c) atomicAdd(&g_counts[i], c);
    }
}

// ---------------------------------------------------------------------------
// Kernel 3: perplexity / utilization from the 1024-bin histogram
// ---------------------------------------------------------------------------
__global__ __launch_bounds__(1024) void fsq_finalize(
    const unsigned* __restrict__ counts,
    float* __restrict__ scalars,   // [commit, codebook, perplexity, utilization]
    float invB)
{
    __shared__ float s_ent[FSQ_NUM_CODES];
    __shared__ float s_used[FSQ_NUM_CODES];
    const int t = threadIdx.x;
    unsigned c = counts[t];
    float p = (float)c * invB;
    s_ent[t]  = p * logf(p + 1e-10f);
    s_used[t] = (c > 0u) ? 1.f : 0.f;
    __syncthreads();
    #pragma unroll
    for (int s = 512; s > 0; s >>= 1) {
        if (t < s) { s_ent[t] += s_ent[t + s]; s_used[t] += s_used[t + s]; }
        __syncthreads();
    }
    if (t == 0) {
        scalars[0] = 0.f;                          // commitment_loss
        scalars[1] = 0.f;                          // codebook_loss
        scalars[2] = expf(-s_ent[0]);              // perplexity
        scalars[3] = s_used[0] * (1.f / 1024.f);   // utilization
    }
}

// ---------------------------------------------------------------------------
extern "C" void kernel_launch(void* const* d_in, const int* in_sizes, int n_in,
                              void* d_out, int out_size, void* d_ws, size_t ws_size,
                              hipStream_t stream) {
    const float* z_e   = (const float*)d_in[0];   // (B, 64)
    const float* w_in  = (const float*)d_in[1];   // (4, 64)
    const float* w_out = (const float*)d_in[2];   // (64, 4)

    const long long nRows = (long long)in_sizes[0] / FSQ_B_LATENT;  // B
    float* zq_out  = (float*)d_out;
    float* idx_out = zq_out + nRows * FSQ_B_LATENT;
    float* scalars = idx_out + nRows;

    unsigned* counts = (unsigned*)d_ws;           // 1024 * 4 bytes of scratch

    fsq_zero_counts<<<(FSQ_NUM_CODES + 255) / 256, 256, 0, stream>>>(counts);

    const int nBlocks = (int)(nRows / FSQ_ROWS);  // B = 524288 -> 2048 blocks
    fsq_main<<<nBlocks, 256, 0, stream>>>(z_e, w_in, w_out, zq_out, idx_out, counts);

    fsq_finalize<<<1, 1024, 0, stream>>>(counts, scalars, 1.f / (float)nRows);
}